// GATLayer_11931419149195
// MI455X (gfx1250) — compile-verified
//
#include <hip/hip_runtime.h>
#include <math.h>

// MI455X / gfx1250: wave32 only.
typedef __attribute__((ext_vector_type(2))) float v2f;
typedef __attribute__((ext_vector_type(8))) float v8f;

#define IN_DIM 64
#define HIDDEN 64

// ---------------------------------------------------------------------------
// Init: out = 0, segmax = -inf, segsum = 0
// ---------------------------------------------------------------------------
__global__ void gat_init_kernel(float* __restrict__ out, float* __restrict__ segmax,
                                float* __restrict__ segsum, int n_nodes) {
    int idx = blockIdx.x * blockDim.x + threadIdx.x;
    int total = n_nodes * HIDDEN;
    if (idx < total) out[idx] = 0.0f;
    if (idx < n_nodes) {
        segmax[idx] = -INFINITY;
        segsum[idx] = 0.0f;
    }
}

// ---------------------------------------------------------------------------
// keys = inputs @ W1 + b1  via V_WMMA_F32_16X16X4_F32 (fp32 matrix cores).
// One wave computes a 16x64 output tile: 4 accumulators (N-tiles of 16),
// K-loop of 16 steps of K=4.
//
// Fragment layouts (ISA 7.12.2, wave32):
//   A 16x4:  lane l, half=l>>4, m=l&15 : v0 = A[m][k0+2*half], v1 = A[m][k0+2*half+1]
//   B 4x16:  lane l, half=l>>4, n=l&15 : v0 = B[k0+2*half][n], v1 = B[k0+2*half+1][n]
//   C 16x16: lane l, n=l&15 : d[r] = C[r + 8*half][n]
// ---------------------------------------------------------------------------
__global__ void gat_keys_gemm_wmma(const float* __restrict__ x, const float* __restrict__ W1,
                                   const float* __restrict__ b1, float* __restrict__ keys,
                                   int n_nodes) {
    const int lane = threadIdx.x & 31;
    const int waveInBlock = threadIdx.x >> 5;
    const int tile = blockIdx.x * (blockDim.x >> 5) + waveInBlock;
    const int m0 = tile * 16;
    if (m0 >= n_nodes) return;

    const int half = lane >> 4;  // which half-wave
    const int sub  = lane & 15;  // row (A) / column (B,C,D) within tile

    if (m0 + 16 <= n_nodes) {
        // Full tile: EXEC is all-ones for every WMMA (uniform branch).
        v8f acc0 = {}, acc1 = {}, acc2 = {}, acc3 = {};
        const float* arow = x + (size_t)(m0 + sub) * IN_DIM;

#pragma unroll
        for (int k0 = 0; k0 < IN_DIM; k0 += 4) {
            const int ka = k0 + 2 * half;
            v2f a;
            a.x = arow[ka];
            a.y = arow[ka + 1];
            const float* brow0 = W1 + (size_t)ka * HIDDEN + sub;        // row ka
            const float* brow1 = W1 + (size_t)(ka + 1) * HIDDEN + sub;  // row ka+1
            v2f b;
            b.x = brow0[0];  b.y = brow1[0];
            acc0 = __builtin_amdgcn_wmma_f32_16x16x4_f32(false, a, false, b, (short)0, acc0, false, false);
            b.x = brow0[16]; b.y = brow1[16];
            acc1 = __builtin_amdgcn_wmma_f32_16x16x4_f32(false, a, false, b, (short)0, acc1, false, false);
            b.x = brow0[32]; b.y = brow1[32];
            acc2 = __builtin_amdgcn_wmma_f32_16x16x4_f32(false, a, false, b, (short)0, acc2, false, false);
            b.x = brow0[48]; b.y = brow1[48];
            acc3 = __builtin_amdgcn_wmma_f32_16x16x4_f32(false, a, false, b, (short)0, acc3, false, false);
        }

        const float bia0 = b1[sub];
        const float bia1 = b1[16 + sub];
        const float bia2 = b1[32 + sub];
        const float bia3 = b1[48 + sub];
#pragma unroll
        for (int r = 0; r < 8; ++r) {
            float* orow = keys + (size_t)(m0 + r + 8 * half) * HIDDEN;
            orow[sub]      = acc0[r] + bia0;
            orow[16 + sub] = acc1[r] + bia1;
            orow[32 + sub] = acc2[r] + bia2;
            orow[48 + sub] = acc3[r] + bia3;
        }
    } else {
        // Tail tile (never taken for N=50000, kept for generality; no WMMA here).
        for (int m = m0; m < n_nodes; ++m) {
            for (int c = lane; c < HIDDEN; c += 32) {
                float s = b1[c];
                for (int k = 0; k < IN_DIM; ++k)
                    s += x[(size_t)m * IN_DIM + k] * W1[(size_t)k * HIDDEN + c];
                keys[(size_t)m * HIDDEN + c] = s;
            }
        }
    }
}

// ---------------------------------------------------------------------------
// Float atomic max via signed/unsigned integer atomics (works with -inf init).
// ---------------------------------------------------------------------------
__device__ __forceinline__ void atomicMaxFloat(float* addr, float value) {
    if (value >= 0.0f) {
        atomicMax((int*)addr, __float_as_int(value));
    } else {
        atomicMin((unsigned int*)addr, __float_as_uint(value));
    }
}

// ---------------------------------------------------------------------------
// Per-edge attention logit + segment max. One wave per edge; each lane covers
// 2 of the 64 channels; wave shuffle reduction for the W2 dot product.
// ---------------------------------------------------------------------------
__global__ void gat_edge_logits(const float* __restrict__ keys, const int* __restrict__ src,
                                const int* __restrict__ dst, const int* __restrict__ etype,
                                const float* __restrict__ W2, const float* __restrict__ b2,
                                const float* __restrict__ edge_emb, float* __restrict__ logits,
                                float* __restrict__ segmax, int n_edges) {
    const int lane = threadIdx.x & 31;
    const int e = (blockIdx.x * blockDim.x + threadIdx.x) >> 5;
    if (e >= n_edges) return;

    const int s = src[e], d = dst[e], t = etype[e];
    const float* ks = keys + (size_t)s * HIDDEN;
    const float* kd = keys + (size_t)d * HIDDEN;
    const float* ee = edge_emb + (size_t)t * HIDDEN;

    float p = (ks[lane] + ee[lane]) * W2[lane]
            + (ks[lane + 32] + ee[lane + 32]) * W2[lane + 32]
            + kd[lane] * W2[HIDDEN + lane]
            + kd[lane + 32] * W2[HIDDEN + lane + 32];

#pragma unroll
    for (int off = 16; off > 0; off >>= 1)
        p += __shfl_xor(p, off, 32);

    if (lane == 0) {
        const float xv = p + b2[0];
        const float lg = fmaxf(0.2f * xv, xv);  // leaky_relu
        logits[e] = lg;
        atomicMaxFloat(&segmax[d], lg);
    }
}

// ---------------------------------------------------------------------------
// Fused: el = exp(logit - segmax[d]); segsum[d] += el;
//        out[d][h] += (keys[s][h] + ee[t][h]) * el   (unnormalized scatter)
// Normalizer is constant per destination, so dividing afterwards is exact.
// Targets (out: 12.8 MB, segsum: 200 KB) are L2-resident on the 192 MB L2.
// ---------------------------------------------------------------------------
__global__ void gat_edge_scatter(const float* __restrict__ keys, const int* __restrict__ src,
                                 const int* __restrict__ dst, const int* __restrict__ etype,
                                 const float* __restrict__ edge_emb, const float* __restrict__ logits,
                                 const float* __restrict__ segmax, float* __restrict__ segsum,
                                 float* __restrict__ out, int n_edges) {
    const int lane = threadIdx.x & 31;
    const int e = (blockIdx.x * blockDim.x + threadIdx.x) >> 5;
    if (e >= n_edges) return;

    const int s = src[e], d = dst[e], t = etype[e];
    const float el = expf(logits[e] - segmax[d]);

    const float* ks = keys + (size_t)s * HIDDEN;
    const float* ee = edge_emb + (size_t)t * HIDDEN;
    float* orow = out + (size_t)d * HIDDEN;

    atomicAdd(&orow[lane],      (ks[lane] + ee[lane]) * el);
    atomicAdd(&orow[lane + 32], (ks[lane + 32] + ee[lane + 32]) * el);
    if (lane == 0) atomicAdd(&segsum[d], el);
}

// ---------------------------------------------------------------------------
// out[n][h] /= segsum[n]; empty segments stay 0 (matches reference).
// ---------------------------------------------------------------------------
__global__ void gat_normalize(float* __restrict__ out, const float* __restrict__ segsum,
                              int n_nodes) {
    const int idx = blockIdx.x * blockDim.x + threadIdx.x;
    if (idx >= n_nodes * HIDDEN) return;
    const float s = segsum[idx >> 6];  // HIDDEN == 64
    const float v = out[idx];
    out[idx] = (s > 0.0f) ? (v / s) : 0.0f;
}

// ---------------------------------------------------------------------------
extern "C" void kernel_launch(void* const* d_in, const int* in_sizes, int n_in,
                              void* d_out, int out_size, void* d_ws, size_t ws_size,
                              hipStream_t stream) {
    const float* x    = (const float*)d_in[0];  // [N, 64]
    const int*   src  = (const int*)d_in[1];    // [E]
    const int*   dst  = (const int*)d_in[2];    // [E]
    const int*   et   = (const int*)d_in[3];    // [E]
    // d_in[4] = num_nodes (device scalar); N derived host-side from in_sizes.
    const float* W1   = (const float*)d_in[5];  // [64, 64]
    const float* b1   = (const float*)d_in[6];  // [64]
    const float* W2   = (const float*)d_in[7];  // [128, 1]
    const float* b2   = (const float*)d_in[8];  // [1]
    const float* ee   = (const float*)d_in[9];  // [6, 64]

    const int N = in_sizes[0] / IN_DIM;  // 50000
    const int E = in_sizes[1];           // 800000

    float* ws     = (float*)d_ws;
    float* keys   = ws;                          // N*64 floats (12.8 MB)
    float* logits = keys + (size_t)N * HIDDEN;   // E floats   (3.2 MB)
    float* segmax = logits + E;                  // N floats
    float* segsum = segmax + N;                  // N floats
    float* out    = (float*)d_out;               // N*64 floats

    const int threads = 256;

    // 1) zero output, init segment reductions
    {
        int total = N * HIDDEN;
        gat_init_kernel<<<(total + threads - 1) / threads, threads, 0, stream>>>(
            out, segmax, segsum, N);
    }
    // 2) keys = x @ W1 + b1 (WMMA f32)
    {
        int tiles = (N + 15) / 16;          // 3125, all full (50000 % 16 == 0)
        int wavesPerBlock = threads / 32;   // 8
        gat_keys_gemm_wmma<<<(tiles + wavesPerBlock - 1) / wavesPerBlock, threads, 0, stream>>>(
            x, W1, b1, keys, N);
    }
    // 3) per-edge logits + segment max (one wave per edge)
    {
        long long tthreads = (long long)E * 32;
        gat_edge_logits<<<(int)((tthreads + threads - 1) / threads), threads, 0, stream>>>(
            keys, src, dst, et, W2, b2, ee, logits, segmax, E);
    }
    // 4) fused exp + unnormalized scatter + segment sum
    {
        long long tthreads = (long long)E * 32;
        gat_edge_scatter<<<(int)((tthreads + threads - 1) / threads), threads, 0, stream>>>(
            keys, src, dst, et, ee, logits, segmax, segsum, out, E);
    }
    // 5) normalize
    {
        int total = N * HIDDEN;
        gat_normalize<<<(total + threads - 1) / threads, threads, 0, stream>>>(out, segsum, N);
    }
}